// SelfCorrelationPercPooling_20976620273817
// MI455X (gfx1250) — compile-verified
//
#include <hip/hip_runtime.h>
#include <hip/hip_bf16.h>
#include <math.h>

// SelfCorrelationPercPooling, fused for MI455X (gfx1250, wave32).
//  B=8, C=512, H=W=48, M=H*W=2304, NB_POOLS=256
//  Per workgroup: one 16-row strip of G = A*A^T/C via V_WMMA_F32_16X16X4_F32,
//  strip kept in LDS (288KB of the 320KB/WGP budget), bitonic-sorted
//  descending, 256 order statistics emitted. No d_ws needed.

typedef __attribute__((ext_vector_type(2))) float v2f;
typedef __attribute__((ext_vector_type(8))) float v8f;

namespace scpp {
constexpr int B_DIM = 8;
constexpr int C_DIM = 512;      // nb_feats (K of the GEMM)
constexpr int M_DIM = 2304;     // H*W = nb_maps
constexpr int NSORT = 4096;     // M padded to power of two for bitonic sort
constexpr int POOLS = 256;
constexpr int TILES_M = M_DIM / 16;        // 144 column tiles of 16
constexpr int WAVES = 8;                   // 256 threads / wave32
constexpr int TILES_PER_WAVE = TILES_M / WAVES;  // 18
}

__global__ __launch_bounds__(256) void scpp_kernel(const float* __restrict__ X,
                                                   float* __restrict__ out) {
  using namespace scpp;
  extern __shared__ float smem[];
  float* As = smem;                    // 16 x 512  A-strip staging (32 KB)
  float* Gs = smem + 16 * C_DIM;       // 16 x 4096 sort buffer     (256 KB)

  const int tid = threadIdx.x;
  const int blk = blockIdx.x;
  const int b   = blk / TILES_M;
  const int m0  = (blk % TILES_M) * 16;      // strip row base in G

  const float* Xb = X + (size_t)b * M_DIM * C_DIM;   // batch as [2304 x 512]

  // ---- stage A strip: rows m0..m0+15 are 8192 contiguous floats ----------
  {
    const float4* src = (const float4*)(Xb + (size_t)m0 * C_DIM);
    float4* dst = (float4*)As;
    for (int i = tid; i < (16 * C_DIM) / 4; i += 256) dst[i] = src[i];
  }
  // ---- -inf padding for the sort region (cols 2304..4095) ----------------
  for (int i = tid; i < 16 * (NSORT - M_DIM); i += 256) {
    int r = i / (NSORT - M_DIM);
    int c = M_DIM + (i % (NSORT - M_DIM));
    Gs[r * NSORT + c] = -__builtin_inff();
  }
  __syncthreads();

  // ---- GEMM: G[m0:m0+16, :] = A_strip * A^T / C via WMMA f32 16x16x4 -----
  // Per-lane operand gather (both A and B operands of A*A^T share the form):
  //   elem[j] = A[rowbase + lane%16][kb + (lane>>4)*2 + j]
  const int lane   = tid & 31;
  const int wave   = tid >> 5;
  const int rowSel = lane & 15;
  const int kSub   = (lane >> 4) * 2;
  const float scale = 1.0f / (float)C_DIM;

  const float* aPtr = As + rowSel * C_DIM + kSub;    // LDS, reused 18x/wave

  for (int t = 0; t < TILES_PER_WAVE; ++t) {
    const int n0 = (wave * TILES_PER_WAVE + t) * 16;
    const float* bPtr = Xb + (size_t)(n0 + rowSel) * C_DIM + kSub;  // L2-hot
    if (t + 1 < TILES_PER_WAVE)
      __builtin_prefetch(Xb + (size_t)(n0 + 16 + rowSel) * C_DIM, 0, 3);

    v8f acc = {0.f, 0.f, 0.f, 0.f, 0.f, 0.f, 0.f, 0.f};
#pragma unroll 4
    for (int kb = 0; kb < C_DIM; kb += 4) {
      v2f a  = *(const v2f*)(aPtr + kb);
      v2f bb = *(const v2f*)(bPtr + kb);
      // 8 args: (neg_a, A, neg_b, B, c_mod, C, reuse_a, reuse_b)
      acc = __builtin_amdgcn_wmma_f32_16x16x4_f32(false, a, false, bb,
                                                  (short)0, acc, false, false);
    }
    // c[v] -> G[m0 + v + 8*(lane>=16)][n0 + lane%16]
    const int mRow = (lane >> 4) * 8;
#pragma unroll
    for (int v = 0; v < 8; ++v)
      Gs[(mRow + v) * NSORT + n0 + rowSel] = acc[v] * scale;
  }
  __syncthreads();

  // ---- bitonic sort: 16 rows x 4096 (descending), all rows in parallel ---
  for (int k = 2; k <= NSORT; k <<= 1) {
    for (int j = k >> 1; j > 0; j >>= 1) {
      for (int e = tid; e < 16 * NSORT; e += 256) {
        int r = e >> 12;              // NSORT == 2^12
        int i = e & (NSORT - 1);
        int l = i ^ j;
        if (l > i) {
          float va = Gs[r * NSORT + i];
          float vb = Gs[r * NSORT + l];
          bool descSeg = ((i & k) == 0);
          if (descSeg ? (va < vb) : (va > vb)) {
            Gs[r * NSORT + i] = vb;
            Gs[r * NSORT + l] = va;
          }
        }
      }
      __syncthreads();
    }
  }

  // ---- gather order statistics; lanes 0..15 write 16 consecutive addrs ---
  // ranks[p] = rint(1 + p*(2302/255)) matches np.round(linspace(1, M-1, 256))
  const float step = 2302.0f / 255.0f;
  const int r  = tid & 15;
  const int mG = m0 + r;
  for (int p = tid >> 4; p < POOLS; p += 16) {
    int rank = (int)rintf(1.0f + (float)p * step);
    out[((size_t)(b * POOLS + p)) * M_DIM + mG] = Gs[r * NSORT + rank];
  }
}

extern "C" void kernel_launch(void* const* d_in, const int* in_sizes, int n_in,
                              void* d_out, int out_size, void* d_ws, size_t ws_size,
                              hipStream_t stream) {
  using namespace scpp;
  (void)in_sizes; (void)n_in; (void)d_ws; (void)ws_size; (void)out_size;
  const float* X = (const float*)d_in[0];
  float* out = (float*)d_out;

  const size_t shmem = (size_t)(16 * C_DIM + 16 * NSORT) * sizeof(float); // 288 KB
  // CDNA5 allows up to 320KB LDS per workgroup; raise the HIP-side cap.
  (void)hipFuncSetAttribute(reinterpret_cast<const void*>(scpp_kernel),
                            hipFuncAttributeMaxDynamicSharedMemorySize,
                            (int)shmem);

  dim3 grid(B_DIM * TILES_M);   // 8 * 144 = 1152 workgroups
  dim3 block(256);              // 8 wave32s
  scpp_kernel<<<grid, block, shmem, stream>>>(X, out);
}